// smoothcrossentropy_22935125361002
// MI455X (gfx1250) — compile-verified
//
#include <hip/hip_runtime.h>

// Smooth cross-entropy: loss = -(1/(B*C)) * sum_r sum_k kw[k]*log(pred[r, label[r]+k-2] + 1e-8)
// B = 262144 rows, C = 121 classes, 5-tap kernel weights.
// Memory-bound gather (~24B useful per row). Deterministic 2-pass reduction;
// cross-lane (wave32) reduction done with V_WMMA_F32_16X16X4_F32 (fp32, fixed order).

#define NUM_CLASS   121
#define NROWS       262144
#define BLOCK_THR   256
#define NBLOCKS     512          // 512*256 threads * 2 rows/thread = 262144 rows
#define WAVES_PER_BLOCK (BLOCK_THR / 32)

typedef __attribute__((ext_vector_type(2))) float v2f;
typedef __attribute__((ext_vector_type(8))) float v8f;

// Full 32-lane sum of `acc` using V_WMMA_F32_16X16X4_F32.
// Step 1: A(16x4) = {acc, 0} per lane  (lanes 0-15 -> K=0, lanes 16-31 -> K=2),
//         B = ones(4x16)  =>  D1[m,n] = acc[m] + acc[m+16]  (= s[m], all n equal).
// Step 2: D1's 8 VGPRs hold matrix rows broadcast across lanes. Feeding VGPR
//         pairs (v0,v1),(v2,v3),(v4,v5),(v6,v7) back as 16x4 A-matrices with
//         B = ones and chaining C accumulates s[0..15] into every element:
//         pair(v0,v1) contributes s0+s1+s8+s9, pair(v2,v3) -> s2+s3+s10+s11, etc.
// Result: every element of final D is the full wave sum; return element 0.
// EXEC must be all 1s at the call site (uniform control flow, full waves).
__device__ __forceinline__ float wave_reduce_sum_wmma(float acc) {
  v2f ones; ones[0] = 1.0f; ones[1] = 1.0f;
  v2f a;    a[0] = acc;     a[1] = 0.0f;
  v8f zero = {0.f, 0.f, 0.f, 0.f, 0.f, 0.f, 0.f, 0.f};

  v8f d1 = __builtin_amdgcn_wmma_f32_16x16x4_f32(false, a, false, ones,
                                                 (short)0, zero, false, false);
  v8f c = zero;
  v2f a2;
  a2[0] = d1[0]; a2[1] = d1[1];
  c = __builtin_amdgcn_wmma_f32_16x16x4_f32(false, a2, false, ones, (short)0, c, false, false);
  a2[0] = d1[2]; a2[1] = d1[3];
  c = __builtin_amdgcn_wmma_f32_16x16x4_f32(false, a2, false, ones, (short)0, c, false, false);
  a2[0] = d1[4]; a2[1] = d1[5];
  c = __builtin_amdgcn_wmma_f32_16x16x4_f32(false, a2, false, ones, (short)0, c, false, false);
  a2[0] = d1[6]; a2[1] = d1[7];
  c = __builtin_amdgcn_wmma_f32_16x16x4_f32(false, a2, false, ones, (short)0, c, false, false);
  return c[0];
}

__global__ __launch_bounds__(BLOCK_THR)
void sce_partial_kernel(const float* __restrict__ pred,
                        const int*   __restrict__ label,
                        const float* __restrict__ kw,
                        float*       __restrict__ partial) {
  __shared__ float s_wave[WAVES_PER_BLOCK];

  // 5-tap weights: tiny, L2/L0 resident after first block.
  float w[5];
#pragma unroll
  for (int k = 0; k < 5; ++k) w[k] = kw[k];

  const int tid      = blockIdx.x * BLOCK_THR + threadIdx.x;
  const int nthreads = NBLOCKS * BLOCK_THR;

  float acc = 0.0f;  // accumulates +kw[k]*log(...); negate once at the end
  // NROWS divisible by nthreads -> uniform trip count -> EXEC stays all-1s.
  for (int r = tid; r < NROWS; r += nthreads) {
    const int lab = label[r];
    const float* __restrict__ row = pred + (size_t)r * NUM_CLASS;
#pragma unroll
    for (int k = 0; k < 5; ++k) {
      const int c = lab - 2 + k;
      if ((unsigned)c < (unsigned)NUM_CLASS) {
        acc += w[k] * __logf(row[c] + 1e-8f);
      }
    }
  }

  // Wave-level fp32 reduction via WMMA (deterministic).
  const float wsum = wave_reduce_sum_wmma(acc);

  const int lane = threadIdx.x & 31;
  const int wave = threadIdx.x >> 5;
  if (lane == 0) s_wave[wave] = wsum;
  __syncthreads();

  if (threadIdx.x == 0) {
    float bsum = 0.0f;
#pragma unroll
    for (int i = 0; i < WAVES_PER_BLOCK; ++i) bsum += s_wave[i];  // fixed order
    partial[blockIdx.x] = bsum;
  }
}

__global__ __launch_bounds__(32)
void sce_final_kernel(const float* __restrict__ partial,
                      float*       __restrict__ out) {
  float acc = 0.0f;
#pragma unroll
  for (int i = (int)threadIdx.x; i < NBLOCKS; i += 32) acc += partial[i];  // fixed order
  const float total = wave_reduce_sum_wmma(acc);
  if (threadIdx.x == 0) {
    // mean over B*C elements, with the leading minus sign.
    out[0] = -total * (1.0f / ((float)NROWS * (float)NUM_CLASS));
  }
}

extern "C" void kernel_launch(void* const* d_in, const int* in_sizes, int n_in,
                              void* d_out, int out_size, void* d_ws, size_t ws_size,
                              hipStream_t stream) {
  const float* pred   = (const float*)d_in[0];   // (262144, 121) f32
  const int*   label  = (const int*)d_in[1];     // (262144,) int
  const float* kweight= (const float*)d_in[2];   // (5,) f32
  float* out     = (float*)d_out;                // scalar mean
  float* partial = (float*)d_ws;                 // NBLOCKS floats of scratch

  sce_partial_kernel<<<NBLOCKS, BLOCK_THR, 0, stream>>>(pred, label, kweight, partial);
  sce_final_kernel<<<1, 32, 0, stream>>>(partial, out);
}